// EMA_11098195492985
// MI455X (gfx1250) — compile-verified
//
#include <hip/hip_runtime.h>
#include <hip/hip_bf16.h>
#include <math.h>

// Shapes: b=16, c=512, h=w=64, GROUPS=8 -> bg=128, cg=64, half=32, HW=4096, L=128
#define BG    128
#define CG    64
#define HWPX  4096
#define LDIM  128
#define EPSV  1e-5f

typedef __attribute__((ext_vector_type(2))) float v2f;
typedef __attribute__((ext_vector_type(8))) float v8f;

__device__ __forceinline__ float sigmoidf(float x) {
    return 1.0f / (1.0f + expf(-x));
}

// -------------------------------------------------------------------------
// K1: per (g,c) slice of 4096: row means -> hwcat[0..63], col means -> hwcat[64..127],
//     plus sum and sumsq of the slice.
// -------------------------------------------------------------------------
__global__ __launch_bounds__(256) void k1_reduce(const float* __restrict__ x,
                                                 float* __restrict__ hwcat,
                                                 float* __restrict__ sum_,
                                                 float* __restrict__ sumsq_) {
    int slice = blockIdx.x;                       // g*64 + c
    const float* base = x + (size_t)slice * HWPX;
    __shared__ float sl[HWPX];
    __shared__ float red[256];
    __shared__ float red2[256];
    int tid = threadIdx.x;
    if (tid == 0) __builtin_prefetch(base + HWPX, 0, 0);   // global_prefetch_b8
    float s = 0.f, s2 = 0.f;
    const float4* b4 = (const float4*)base;
    float4* sl4 = (float4*)sl;
    #pragma unroll
    for (int i = 0; i < 4; ++i) {
        int v = tid + i * 256;
        float4 d = b4[v];
        sl4[v] = d;
        s  += d.x + d.y + d.z + d.w;
        s2 += d.x*d.x + d.y*d.y + d.z*d.z + d.w*d.w;
    }
    red[tid] = s; red2[tid] = s2;
    __syncthreads();
    for (int off = 128; off > 0; off >>= 1) {
        if (tid < off) { red[tid] += red[tid+off]; red2[tid] += red2[tid+off]; }
        __syncthreads();
    }
    if (tid == 0) { sum_[slice] = red[0]; sumsq_[slice] = red2[0]; }
    if (tid < 64) {
        int h = tid;
        float rs = 0.f;
        #pragma unroll 8
        for (int j = 0; j < 64; ++j) { int w = (h + j) & 63; rs += sl[h*64 + w]; }
        hwcat[(size_t)slice * LDIM + h] = rs * (1.0f/64.0f);
        int w = tid;
        float cs = 0.f;
        #pragma unroll 8
        for (int j = 0; j < 64; ++j) cs += sl[j*64 + w];
        hwcat[(size_t)slice * LDIM + 64 + w] = cs * (1.0f/64.0f);
    }
}

// -------------------------------------------------------------------------
// K2: per group g: hw = W(64x64) @ hwcat(64x128) + bias ; store sigmoid(hw).
// fp32 WMMA 16x16x4. 4 waves, wave wv owns M rows [16wv, 16wv+16).
// A frag (16x4 f32): lanes 0-15 hold row M=l, VGPR0=K(+0|+2 by half), VGPR1=K+1|+3.
// B frag (4x16 f32): lanes hold col N=l, same K striping.
// D (16x16 f32): VGPR r = row r (lanes 0-15) / row 8+r (lanes 16-31).
// -------------------------------------------------------------------------
__global__ __launch_bounds__(128) void k2_gemm(const float* __restrict__ Wm,
                                               const float* __restrict__ bias,
                                               const float* __restrict__ hwcat,
                                               float* __restrict__ sgm) {
    int g = blockIdx.x;
    __shared__ float Wl[64*64];
    __shared__ float Hl[64*128];
    int tid = threadIdx.x;
    {
        const float4* s = (const float4*)Wm;
        float4* d = (float4*)Wl;
        for (int i = tid; i < 1024; i += 128) d[i] = s[i];
        const float4* s2 = (const float4*)(hwcat + (size_t)g * (CG*LDIM));
        float4* d2 = (float4*)Hl;
        for (int i = tid; i < 2048; i += 128) d2[i] = s2[i];
    }
    __syncthreads();
    int wv   = tid >> 5;        // 0..3  (M tile)
    int lane = tid & 31;
    int hf   = lane >> 4;       // half-wave
    int l    = lane & 15;
    int row0 = wv * 16;
    for (int n = 0; n < 8; ++n) {
        v8f acc = {0.f,0.f,0.f,0.f,0.f,0.f,0.f,0.f};
        #pragma unroll
        for (int kk = 0; kk < 16; ++kk) {
            int k0 = kk*4 + 2*hf;
            v2f a, bf;
            a.x  = Wl[(row0 + l)*64 + k0];
            a.y  = Wl[(row0 + l)*64 + k0 + 1];
            bf.x = Hl[(k0    )*128 + n*16 + l];
            bf.y = Hl[(k0 + 1)*128 + n*16 + l];
            acc = __builtin_amdgcn_wmma_f32_16x16x4_f32(
                false, a, false, bf, (short)0, acc, false, false);
        }
        #pragma unroll
        for (int r = 0; r < 8; ++r) {
            int row = row0 + r + 8*hf;
            int col = n*16 + l;
            float v = sigmoidf(acc[r] + bias[row]);
            sgm[(size_t)g * (CG*LDIM) + (size_t)row * LDIM + col] = v;
        }
    }
}

// -------------------------------------------------------------------------
// K3: per (g,c): y = gx * sig(xh[h]) * sig(xw[w]); accumulate sum(y), sum(y^2).
// For c>=32 (i=c-32): also sum over pixels of gx * sigmoid(sw_i*xs + sb_i),
// where xs = gx*as + bs from the gx stats (GN1 affine).
// -------------------------------------------------------------------------
__global__ __launch_bounds__(256) void k3_ystats(const float* __restrict__ x,
                                                 const float* __restrict__ sgm,
                                                 const float* __restrict__ sum_,
                                                 const float* __restrict__ sumsq_,
                                                 const float* __restrict__ gn1_w,
                                                 const float* __restrict__ gn1_b,
                                                 const float* __restrict__ sweight,
                                                 const float* __restrict__ sbias,
                                                 float* __restrict__ sumy,
                                                 float* __restrict__ sumy2,
                                                 float* __restrict__ ssp) {
    int slice = blockIdx.x;
    int g = slice >> 6;
    int c = slice & 63;
    __shared__ float sigh[64], sigw[64];
    __shared__ float asbs[2];
    __shared__ float red[256], red2[256], red3[256];
    int tid = threadIdx.x;
    if (tid < 128) {
        float v = sgm[(size_t)slice * LDIM + tid];
        if (tid < 64) sigh[tid] = v; else sigw[tid-64] = v;
    }
    if (tid == 0) {
        if (c >= 32) {
            int i = c - 32;
            float m   = sum_[slice]   * (1.f/4096.f);
            float var = sumsq_[slice] * (1.f/4096.f) - m*m;
            float ri  = rsqrtf(var + EPSV);
            float a   = ri * gn1_w[i];
            asbs[0] = a; asbs[1] = gn1_b[i] - m*a;
        } else { asbs[0] = 0.f; asbs[1] = 0.f; }
        __builtin_prefetch(x + (size_t)(slice+1) * HWPX, 0, 0);
    }
    __syncthreads();
    float as = asbs[0], bs = asbs[1];
    float swv = 0.f, sbv = 0.f;
    if (c >= 32) { swv = sweight[c-32]; sbv = sbias[c-32]; }
    int h = tid >> 2, wq = tid & 3;
    float sh = sigh[h];
    float sy = 0.f, sy2 = 0.f, sp = 0.f;
    const float4* b4 = (const float4*)(x + (size_t)slice * HWPX);
    #pragma unroll
    for (int q = 0; q < 4; ++q) {
        int vidx = h*16 + wq*4 + q;           // p = h*64 + wq*16 + q*4
        float4 d = b4[vidx];
        float vv[4] = {d.x, d.y, d.z, d.w};
        #pragma unroll
        for (int j = 0; j < 4; ++j) {
            int w = wq*16 + q*4 + j;
            float v = vv[j];
            float y = v * sh * sigw[w];
            sy  += y;
            sy2 += y*y;
            float xs = v*as + bs;
            sp += v * sigmoidf(swv*xs + sbv);
        }
    }
    red[tid] = sy; red2[tid] = sy2; red3[tid] = sp;
    __syncthreads();
    for (int off = 128; off > 0; off >>= 1) {
        if (tid < off) { red[tid]+=red[tid+off]; red2[tid]+=red2[tid+off]; red3[tid]+=red3[tid+off]; }
        __syncthreads();
    }
    if (tid == 0) {
        sumy[slice] = red[0]; sumy2[slice] = red2[0];
        if (c >= 32) ssp[g*32 + (c-32)] = red3[0];
    }
}

// -------------------------------------------------------------------------
// K4: per group: softmaxes + per-channel coefficients.
//   mean(x1)[c] = gn_b[c] exactly (GN output mean == bias).
//   mean(x2)[2i]   = xc_i * sig(cw_i*xc_i + cb_i);  mean(x2)[2i+1] = ssp_i/4096.
// coef layout per g (stride 256):
//   [0..63]  t1 = x21[c]*a1[c]            (multiplies y)
//   [64..127] t2: cs<32 -> x11[2cs]*sc_cs (multiplies gx); cs>=32 -> x11[2(cs-32)+1]
//   [128..159] as, [160..191] bs (GN1 affine), [192] C = sum x21[c]*b1[c]
// -------------------------------------------------------------------------
__device__ __forceinline__ float softmax64(float val, float* red, int c) {
    red[c] = val; __syncthreads();
    for (int off = 32; off > 0; off >>= 1) {
        if (c < off) red[c] = fmaxf(red[c], red[c+off]);
        __syncthreads();
    }
    float mx = red[0]; __syncthreads();
    float e = expf(val - mx);
    red[c] = e; __syncthreads();
    for (int off = 32; off > 0; off >>= 1) {
        if (c < off) red[c] += red[c+off];
        __syncthreads();
    }
    float s = red[0]; __syncthreads();
    return e / s;
}

__global__ __launch_bounds__(64) void k4_coef(const float* __restrict__ sum_,
                                              const float* __restrict__ sumsq_,
                                              const float* __restrict__ sumy,
                                              const float* __restrict__ sumy2,
                                              const float* __restrict__ ssp,
                                              const float* __restrict__ gn_w,
                                              const float* __restrict__ gn_b,
                                              const float* __restrict__ gn1_w,
                                              const float* __restrict__ gn1_b,
                                              const float* __restrict__ cweight,
                                              const float* __restrict__ cbias,
                                              float* __restrict__ coef) {
    int g = blockIdx.x;
    int c = threadIdx.x;              // 0..63
    __shared__ float red[64];
    __shared__ float x11s[64];
    float my = sumy[g*64+c]  * (1.f/4096.f);
    float vy = sumy2[g*64+c] * (1.f/4096.f) - my*my;
    float ri = rsqrtf(vy + EPSV);
    float a1 = ri * gn_w[c];
    float b1 = gn_b[c] - my * a1;
    float mean1 = gn_b[c];
    float mean2;
    if ((c & 1) == 0) {
        int i = c >> 1;
        float xc = sum_[g*64 + i] * (1.f/4096.f);
        mean2 = xc * sigmoidf(cweight[i]*xc + cbias[i]);
    } else {
        int i = (c - 1) >> 1;
        mean2 = ssp[g*32 + i] * (1.f/4096.f);
    }
    float x11 = softmax64(mean1, red, c);
    float x21 = softmax64(mean2, red, c);
    x11s[c] = x11;
    __syncthreads();
    float* out = coef + (size_t)g * 256;
    out[c] = x21 * a1;                               // t1
    float t2;
    if (c < 32) {
        float xc = sum_[g*64 + c] * (1.f/4096.f);
        float sc = sigmoidf(cweight[c]*xc + cbias[c]);
        t2 = x11s[2*c] * sc;
    } else {
        int i = c - 32;
        t2 = x11s[2*i + 1];
    }
    out[64 + c] = t2;
    if (c < 32) {
        int sl = g*64 + 32 + c;
        float m   = sum_[sl]   * (1.f/4096.f);
        float var = sumsq_[sl] * (1.f/4096.f) - m*m;
        float r2  = rsqrtf(var + EPSV);
        float a   = r2 * gn1_w[c];
        out[128 + c] = a;
        out[160 + c] = gn1_b[c] - m*a;
    }
    red[c] = x21 * b1;
    __syncthreads();
    for (int off = 32; off > 0; off >>= 1) {
        if (c < off) red[c] += red[c+off];
        __syncthreads();
    }
    if (c == 0) out[192] = red[0];
}

// -------------------------------------------------------------------------
// K5: fused final pass. Tile = (group g, 512 pixels = 8 image rows).
// Stage all 64 channels of the tile in LDS (64*512*4B = 128 KB, legal on
// CDNA5's 320 KB LDS), compute per-pixel weight = C + sum_cs gx*(t1*sh*sw + ...)
// then out = gx * sigmoid(weight) straight from LDS. One HBM read + one write.
// -------------------------------------------------------------------------
#define TP 512
__global__ __launch_bounds__(256) void k5_final(const float* __restrict__ x,
                                                const float* __restrict__ sgm,
                                                const float* __restrict__ coef,
                                                const float* __restrict__ sweight,
                                                const float* __restrict__ sbias,
                                                float* __restrict__ out) {
    int g    = blockIdx.x >> 3;
    int tile = blockIdx.x & 7;
    int p0   = tile * TP;
    int h0   = p0 >> 6;                // 8 rows per tile
    __shared__ float gxt[64 * TP];     // 128 KB
    __shared__ float sighT[64 * 8];
    __shared__ float sigwT[64 * 64];
    __shared__ float t1s[64], t2s[64], ass[32], bss[32], sws[32], sbss[32];
    __shared__ float Cs;
    int tid = threadIdx.x;
    const float* cf = coef + (size_t)g * 256;
    if (tid < 64) { t1s[tid] = cf[tid]; t2s[tid] = cf[64 + tid]; }
    else if (tid < 96) {
        int i = tid - 64;
        ass[i] = cf[128 + i]; bss[i] = cf[160 + i];
        sws[i] = sweight[i];  sbss[i] = sbias[i];
    } else if (tid == 96) Cs = cf[192];
    for (int i = tid; i < 512; i += 256) {
        int c = i >> 3, hr = i & 7;
        sighT[i] = sgm[(size_t)g * (CG*LDIM) + c*LDIM + h0 + hr];
    }
    for (int i = tid; i < 4096; i += 256) {
        int c = i >> 6, w = i & 63;
        sigwT[i] = sgm[(size_t)g * (CG*LDIM) + c*LDIM + 64 + w];
    }
    const float4* x4 = (const float4*)(x + (size_t)g * CG * HWPX);
    float4* g4 = (float4*)gxt;
    for (int i = tid; i < 64 * TP / 4; i += 256) {
        int c = i / (TP/4);
        int o = i % (TP/4);
        g4[i] = x4[(size_t)c * (HWPX/4) + (p0 >> 2) + o];
    }
    __syncthreads();
    float wsig[2];
    #pragma unroll
    for (int k = 0; k < 2; ++k) {
        int pl = tid + k*256;
        int hl = pl >> 6;
        int w  = pl & 63;
        float acc = Cs;
        #pragma unroll 4
        for (int cs = 0; cs < 32; ++cs) {
            float v = gxt[cs*TP + pl];
            acc += v * (t1s[cs] * sighT[cs*8 + hl] * sigwT[cs*64 + w] + t2s[cs]);
        }
        #pragma unroll 4
        for (int cs = 32; cs < 64; ++cs) {
            int i = cs - 32;
            float v = gxt[cs*TP + pl];
            float xs = v*ass[i] + bss[i];
            acc += v * (t1s[cs] * sighT[cs*8 + hl] * sigwT[cs*64 + w]
                        + t2s[cs] * sigmoidf(sws[i]*xs + sbss[i]));
        }
        wsig[k] = sigmoidf(acc);
    }
    float* ob = out + (size_t)g * CG * HWPX + p0;
    #pragma unroll 4
    for (int c = 0; c < 64; ++c) {
        ob[(size_t)c * HWPX + tid]       = gxt[c*TP + tid]       * wsig[0];
        ob[(size_t)c * HWPX + tid + 256] = gxt[c*TP + tid + 256] * wsig[1];
    }
}

// -------------------------------------------------------------------------
extern "C" void kernel_launch(void* const* d_in, const int* in_sizes, int n_in,
                              void* d_out, int out_size, void* d_ws, size_t ws_size,
                              hipStream_t stream) {
    const float* x       = (const float*)d_in[0];
    const float* conv_w  = (const float*)d_in[1];
    const float* conv_b  = (const float*)d_in[2];
    const float* gn_w    = (const float*)d_in[3];
    const float* gn_b    = (const float*)d_in[4];
    const float* gn1_w   = (const float*)d_in[5];
    const float* gn1_b   = (const float*)d_in[6];
    const float* cweight = (const float*)d_in[7];
    const float* cbias   = (const float*)d_in[8];
    const float* sweight = (const float*)d_in[9];
    const float* sbias   = (const float*)d_in[10];
    float* outp = (float*)d_out;

    float* ws     = (float*)d_ws;
    float* hwcat  = ws;                        // 128*64*128 = 1,048,576
    float* sgm    = hwcat + 1048576;           // 1,048,576
    float* sum_   = sgm + 1048576;             // 8192
    float* sumsq_ = sum_ + 8192;               // 8192
    float* sumy   = sumsq_ + 8192;             // 8192
    float* sumy2  = sumy + 8192;               // 8192
    float* sspv   = sumy2 + 8192;              // 4096
    float* coef   = sspv + 4096;               // 128*256 = 32768  (total ~8.7 MB)

    k1_reduce<<<dim3(BG*CG), dim3(256), 0, stream>>>(x, hwcat, sum_, sumsq_);
    k2_gemm  <<<dim3(BG),    dim3(128), 0, stream>>>(conv_w, conv_b, hwcat, sgm);
    k3_ystats<<<dim3(BG*CG), dim3(256), 0, stream>>>(x, sgm, sum_, sumsq_,
                                                     gn1_w, gn1_b, sweight, sbias,
                                                     sumy, sumy2, sspv);
    k4_coef  <<<dim3(BG),    dim3(64),  0, stream>>>(sum_, sumsq_, sumy, sumy2, sspv,
                                                     gn_w, gn_b, gn1_w, gn1_b,
                                                     cweight, cbias, coef);
    k5_final <<<dim3(BG*8),  dim3(256), 0, stream>>>(x, sgm, coef, sweight, sbias, outp);
}